// HDP_binary_classifier_48567490183198
// MI455X (gfx1250) — compile-verified
//
#include <hip/hip_runtime.h>
#include <stdint.h>

#define DIM       1024
#define NROWS     65536
#define MC_NUM    4096
#define LOG2PI_F  1.8378770664093453f

typedef float v2f __attribute__((ext_vector_type(2)));
typedef float v8f __attribute__((ext_vector_type(8)));

// ---------------- workspace layout (float offsets) ----------------
// B arrays are padded to 16 rows (rows 2..15 zero) so every WMMA lane loads a
// valid row -> no mask VALU, no divergence, dead lanes hit shared zero lines.
#define WS_BA_C   0          // [16][1024]  -0.5/sigma^2   (cluster; rows 2..15 = 0)
#define WS_BB_C   16384      // [16][1024]  mu/sigma^2     (cluster)
#define WS_BA_F   32768      // [16][1024]  (classifier)
#define WS_BB_F   49152      // [16][1024]
#define WS_STD    65536      // [2 sets][2][1024] sigma
#define WS_SCAL   69632      // [0..3] sum log sigma, [4..7] sum mu^2/var,
                             // [8..9] c_c, [10..11] c_f, [12] ent, [13] accum
#define WS_LP_X   69664      // [NROWS][2]
#define WS_LP_MC  200736     // [MC_NUM][2]
#define WS_MC     208928     // [MC_NUM][1024]

// ---------------- stateless RNG ----------------
__device__ __forceinline__ uint32_t hash_u32(uint32_t a, uint32_t b) {
    uint32_t x = a * 0x9E3779B1u + b * 0x85EBCA77u + 0xC2B2AE3Du;
    x ^= x >> 16; x *= 0x7FEB352Du;
    x ^= x >> 15; x *= 0x846CA68Bu;
    x ^= x >> 16;
    return x;
}
__device__ __forceinline__ float uniform01(uint32_t h) {
    return ((float)(h >> 8) + 0.5f) * (1.0f / 16777216.0f);   // (0,1)
}
__device__ __forceinline__ v2f boxmuller(uint32_t h1, uint32_t h2) {
    float u1 = uniform01(h1), u2 = uniform01(h2);
    float r = __fsqrt_rn(-2.0f * __logf(u1));
    float a = 6.28318530717958647f * u2;
    v2f o; o.x = r * __cosf(a); o.y = r * __sinf(a);
    return o;
}

__device__ __forceinline__ float block_reduce_256(float v, float* red) {
    int tid = threadIdx.x;
    red[tid] = v; __syncthreads();
    #pragma unroll
    for (int s = 128; s > 0; s >>= 1) {
        if (tid < s) red[tid] += red[tid + s];
        __syncthreads();
    }
    float r = red[0]; __syncthreads();
    return r;
}

// ---------------- zero-fill padding rows 2..15 of the four B arrays ----------------
__global__ __launch_bounds__(256) void hdp_zero_bpad(float* __restrict__ ws)
{
    // 4 arrays x 14 rows x 1024 floats = 57344 floats = 14336 float4
    const int i = blockIdx.x * 256 + threadIdx.x;      // 0..14335 (grid = 56)
    const int arr = i / 3584;                          // 14*1024/4 = 3584 float4 per array
    const int off = i - arr * 3584;
    float* base = ws + arr * 16384 + 2 * DIM;          // start of row 2
    float4 z = {0.f, 0.f, 0.f, 0.f};
    *(float4*)(base + off * 4) = z;
}

// ---------------- prep: per-cluster coefficient vectors + reductions ----------------
__global__ __launch_bounds__(256) void hdp_prep(
    const float* __restrict__ mu_c, const float* __restrict__ rho_c,
    const float* __restrict__ mu_f, const float* __restrict__ rho_f,
    float* __restrict__ ws)
{
    __shared__ float red[256];
    const int set = blockIdx.x >> 1;     // 0 = cluster, 1 = classifier
    const int t   = blockIdx.x & 1;
    const float* mu  = set ? mu_f  : mu_c;
    const float* rho = set ? rho_f : rho_c;
    float* Ba = ws + (set ? WS_BA_F : WS_BA_C);
    float* Bb = ws + (set ? WS_BB_F : WS_BB_C);
    float* sd = ws + WS_STD + set * 2048;

    float sls = 0.f, smi = 0.f;
    for (int d = threadIdx.x; d < DIM; d += 256) {
        float r  = rho[t * DIM + d];
        float sp = (r > 15.f) ? r : log1pf(__expf(r));   // sigma = log1p(exp(rho))
        float iv = 1.f / (sp * sp);
        float m  = mu[t * DIM + d];
        Ba[t * DIM + d] = -0.5f * iv;
        Bb[t * DIM + d] = m * iv;
        sd[t * DIM + d] = sp;
        sls += __logf(sp);
        smi += m * m * iv;
    }
    float tsls = block_reduce_256(sls, red);
    float tsmi = block_reduce_256(smi, red);
    if (threadIdx.x == 0) {
        ws[WS_SCAL + blockIdx.x]     = tsls;
        ws[WS_SCAL + 4 + blockIdx.x] = tsmi;
    }
}

__global__ void hdp_prep_fin(float* __restrict__ ws)
{
    float* s = ws + WS_SCAL;
    const float D = (float)DIM;
    #pragma unroll
    for (int t = 0; t < 2; ++t) {
        s[8 + t]  = -0.5f * D * LOG2PI_F - s[t]     - 0.5f * s[4 + t];   // c_c
        s[10 + t] = -0.5f * D * LOG2PI_F - s[2 + t] - 0.5f * s[6 + t];   // c_f
    }
    s[12] = 0.5f * D * (LOG2PI_F + 1.f) + 0.5f * (s[0] + s[1]);          // entropy
    s[13] = 0.f;                                                          // accum
}

// ---------------- main WMMA kernel: lp[n][t] = c_t + x^2.Ba_t + x.Bb_t ----------------
// One wave handles a 16-row tile; 512 V_WMMA_F32_16X16X4_F32 per tile.
// x staged through LDS (coalesced float4 loads, pad-68 conflict-free b64 reads);
// B coefficients batch-preloaded into registers per 64-col tile (one wait, then a
// pure ds_load -> v_pk_mul -> wmma,wmma inner loop).
__global__ __launch_bounds__(256) void hdp_quad_wmma(
    const float* __restrict__ X,
    const float* __restrict__ Ba, const float* __restrict__ Bb,  // [16][1024], rows 2+ zero
    const float* __restrict__ cvec,   // [2]
    float* __restrict__ lp)           // [rows][2]
{
    __shared__ float tile[8][16 * 68];           // 34,816 B: per-wave 16x64 tile, stride 68
    const int lane = threadIdx.x & 31;
    const int wave = threadIdx.x >> 5;
    const int lo16 = lane & 15;
    const int hi   = lane >> 4;                  // 0/1: K-half + M-half selector

    const int tidx = blockIdx.x * 8 + wave;      // 16-row tile index (exact coverage)
    const float* xbase = X + (size_t)tidx * 16 * DIM;
    const float* barow = Ba + lo16 * DIM + hi * 2;   // this lane's coefficient row
    const float* bbrow = Bb + lo16 * DIM + hi * 2;

    v8f acc = {0.f, 0.f, 0.f, 0.f, 0.f, 0.f, 0.f, 0.f};

    const int scol = lo16 * 4;                   // staging column for this lane
    for (int ct = 0; ct < DIM / 64; ++ct) {
        // ---- stage 16 rows x 64 cols, fully coalesced ----
        #pragma unroll
        for (int j = 0; j < 8; ++j) {
            const int row = 2 * j + hi;
            const float* src = xbase + (size_t)row * DIM + ct * 64 + scol;
            if (ct + 1 < DIM / 64)
                __builtin_prefetch(src + 64, 0, 1);      // next tile into near cache
            *(float4*)(&tile[wave][row * 68 + scol]) = *(const float4*)(src);
        }

        // ---- batch-preload B operand pairs for this 64-col tile ----
        v2f baq[16], bbq[16];
        #pragma unroll
        for (int ks = 0; ks < 16; ++ks) {
            baq[ks] = *(const v2f*)(barow + ct * 64 + ks * 4);
            bbq[ks] = *(const v2f*)(bbrow + ct * 64 + ks * 4);
        }
        __syncthreads();

        // ---- 16 K-steps of 4; two chained f32 WMMAs per step ----
        #pragma unroll
        for (int ks = 0; ks < 16; ++ks) {
            const int kc = ks * 4 + hi * 2;              // lane's K pair within tile
            v2f xa = *(const v2f*)(&tile[wave][lo16 * 68 + kc]);
            v2f xs = xa * xa;
            acc = __builtin_amdgcn_wmma_f32_16x16x4_f32(
                false, xs, false, baq[ks], (short)0, acc, false, false);
            acc = __builtin_amdgcn_wmma_f32_16x16x4_f32(
                false, xa, false, bbq[ks], (short)0, acc, false, false);
        }
        __syncthreads();
    }

    // D layout: VGPR j -> M=j (lanes 0-15) / M=j+8 (lanes 16-31), N=lane&15
    if (lo16 < 2) {
        const float c = cvec[lo16];
        #pragma unroll
        for (int j = 0; j < 8; ++j) {
            const int row = tidx * 16 + j + hi * 8;
            lp[row * 2 + lo16] = acc[j] + c;
        }
    }
}

// ---------------- likelihood: phi softmax + mean reduction ----------------
__global__ __launch_bounds__(256) void hdp_lik_reduce(
    const float* __restrict__ lp, float* __restrict__ ws)
{
    __shared__ float red[256];
    const int i = blockIdx.x * 256 + threadIdx.x;     // row index (exact: 256*256 = 65536)
    const float ent = ws[WS_SCAL + 12];

    float lp0 = lp[2 * i], lp1 = lp[2 * i + 1];
    float u0 = uniform01(hash_u32(0x7777u, (uint32_t)i));
    float u1 = uniform01(hash_u32(0x8888u, (uint32_t)i));
    float pi0 = u0, pi1 = u1 * (1.f - u0);            // stick-breaking, Beta(1,1)=U(0,1)
    float a0 = __logf(pi0) + ent + lp0;
    float a1 = __logf(pi1) + ent + lp1;
    float mx = fmaxf(a0, a1);
    float e0 = __expf(a0 - mx), e1 = __expf(a1 - mx);
    float inv = 1.f / (e0 + e1);
    float val = (e0 * (ent + lp0) + e1 * (ent + lp1)) * inv;

    float bsum = block_reduce_256(val, red);
    if (threadIdx.x == 0)
        atomicAdd(ws + WS_SCAL + 13, bsum);
}

__global__ void hdp_neglik_fin(const float* __restrict__ ws, float* __restrict__ out)
{
    out[2 * MC_NUM] = -ws[WS_SCAL + 13] * (1.f / (float)NROWS);
}

// ---------------- MC sampling: reparametrized draws from cluster mixture ----------------
__global__ __launch_bounds__(256) void hdp_mc_sample(
    const float* __restrict__ mu_c, const float* __restrict__ ws_std,
    float* __restrict__ mc)
{
    const int gid = blockIdx.x * 256 + threadIdx.x;   // 1,048,576 threads, 4 floats each
    const int m = gid >> 8;
    const int d = (gid & 255) * 4;

    float u0 = uniform01(hash_u32(0x1111u, (uint32_t)m));
    float u1 = uniform01(hash_u32(0x2222u, (uint32_t)m));
    float w0 = u0, w1 = u1 * (1.f - u0);
    float p0 = w0 / (w0 + w1);
    float uc = uniform01(hash_u32(0x3333u, (uint32_t)m));
    const int draw = (uc < p0) ? 0 : 1;

    const uint32_t base = (uint32_t)(m * DIM + d);
    v2f e01 = boxmuller(hash_u32(0x4444u, base),     hash_u32(0x5551u, base));
    v2f e23 = boxmuller(hash_u32(0x4444u, base + 2), hash_u32(0x5551u, base + 2));

    float4 mu = *(const float4*)(mu_c   + draw * DIM + d);
    float4 sd = *(const float4*)(ws_std + draw * DIM + d);
    float4 o;
    o.x = mu.x + sd.x * e01.x;  o.y = mu.y + sd.y * e01.y;
    o.z = mu.z + sd.z * e23.x;  o.w = mu.w + sd.w * e23.y;
    *(float4*)(mc + (size_t)m * DIM + d) = o;
}

// ---------------- classifier head: logits = pi_inf * softmax(lp_inf) ----------------
__global__ __launch_bounds__(256) void hdp_infer_fin(
    const float* __restrict__ lp_mc, float* __restrict__ out)
{
    const int m = blockIdx.x * 256 + threadIdx.x;     // 0..4095
    float lp0 = lp_mc[2 * m], lp1 = lp_mc[2 * m + 1];
    float u0 = uniform01(hash_u32(0xAAAAu, (uint32_t)m));
    float u1 = uniform01(hash_u32(0xBBBBu, (uint32_t)m));
    float pi0 = u0, pi1 = u1 * (1.f - u0);
    float mx = fmaxf(lp0, lp1);
    float e0 = __expf(lp0 - mx), e1 = __expf(lp1 - mx);
    float inv = 1.f / (e0 + e1);
    out[2 * m]     = pi0 * e0 * inv;
    out[2 * m + 1] = pi1 * e1 * inv;
}

// ---------------- host-side orchestration ----------------
extern "C" void kernel_launch(void* const* d_in, const int* in_sizes, int n_in,
                              void* d_out, int out_size, void* d_ws, size_t ws_size,
                              hipStream_t stream) {
    const float* x     = (const float*)d_in[0];
    const float* mu_c  = (const float*)d_in[1];
    const float* rho_c = (const float*)d_in[2];
    const float* mu_f  = (const float*)d_in[3];
    const float* rho_f = (const float*)d_in[4];
    float* out = (float*)d_out;
    float* ws  = (float*)d_ws;

    // 1) coefficient vectors (padded), constants, entropy; zero the accumulator
    hdp_zero_bpad<<<56, 256, 0, stream>>>(ws);
    hdp_prep<<<4, 256, 0, stream>>>(mu_c, rho_c, mu_f, rho_f, ws);
    hdp_prep_fin<<<1, 1, 0, stream>>>(ws);

    // 2) bandwidth-bound WMMA pass over x: 65536/16 = 4096 wave-tiles = 512 blocks
    hdp_quad_wmma<<<512, 256, 0, stream>>>(
        x, ws + WS_BA_C, ws + WS_BB_C, ws + WS_SCAL + 8, ws + WS_LP_X);

    // 3) phi softmax + mean -> neg_likelihood
    hdp_lik_reduce<<<256, 256, 0, stream>>>(ws + WS_LP_X, ws);
    hdp_neglik_fin<<<1, 1, 0, stream>>>(ws, out);

    // 4) MC reparametrized samples [4096,1024]
    hdp_mc_sample<<<4096, 256, 0, stream>>>(mu_c, ws + WS_STD, ws + WS_MC);

    // 5) classifier quad form over MC samples: 4096/16 = 256 wave-tiles = 32 blocks
    hdp_quad_wmma<<<32, 256, 0, stream>>>(
        ws + WS_MC, ws + WS_BA_F, ws + WS_BB_F, ws + WS_SCAL + 10, ws + WS_LP_MC);

    // 6) logits
    hdp_infer_fin<<<16, 256, 0, stream>>>(ws + WS_LP_MC, out);
}